// LowRankEmbedding_19713899888995
// MI455X (gfx1250) — compile-verified
//
#include <hip/hip_runtime.h>
#include <hip/hip_bf16.h>

typedef __attribute__((ext_vector_type(2))) float v2f;
typedef __attribute__((ext_vector_type(8))) float v8f;

#define EMB_DIM 128
#define RANK     32

// out[p, :] = W1[idx[p], :] @ W2      (f32, exact-precision fused low-rank embedding)
// One wave32 handles 16 lookups: A = gathered 16x32 tile of W1, B = W2 (32x128 in LDS,
// transposed), D = 16x128 output rows via 8 col-tiles x 8 K-chunks of
// v_wmma_f32_16x16x4_f32.
//
// Cache policy (MI455X: 192 MB L2, 23.3 TB/s HBM):
//   - W1 (128 MB) fits in L2 and is randomly re-gathered -> default (RT) loads.
//   - Output (419 MB, write-once stream) -> non-temporal stores so it does not
//     evict the hot W1 working set from L2.
//   - Index stream (read-once) -> non-temporal loads.
__global__ __launch_bounds__(256) void LowRankEmbedding_19713899888995_kernel(
    const int*   __restrict__ idx,   // [n_pos] row indices
    const float* __restrict__ W1,    // [NUM_EMB, 32]
    const float* __restrict__ W2,    // [32, 128] row-major
    float*       __restrict__ out,   // [n_pos, 128]
    int n_pos)
{
    // W2 staged transposed: w2t[n][k], so B fragments are contiguous 8B LDS loads.
    __shared__ float w2t[EMB_DIM * RANK];   // 16 KB of the 320 KB/WGP pool

    const int tid = threadIdx.x;
    // 4096 floats / 256 threads = 16 each (coalesced global reads, scalar LDS writes)
    #pragma unroll
    for (int j = 0; j < 16; ++j) {
        int f = tid * 16 + j;       // flat index into row-major W2
        int k = f >> 7;             // f / 128
        int n = f & 127;            // f % 128
        w2t[n * RANK + k] = W2[f];
    }
    __syncthreads();

    const int lane = tid & 31;      // wave32
    const int wave = tid >> 5;      // 0..7
    const int half = lane >> 4;     // 0: lanes 0-15, 1: lanes 16-31
    const int l16  = lane & 15;

    const int tileBase = (blockIdx.x * 8 + wave) * 16;  // first of 16 positions
    if (tileBase >= n_pos) return;  // wave-uniform: EXEC stays all-1s for WMMA

    // Gather one W1 row per M; lanes L and L+16 split the row's K range.
    // A-fragment layout (16x4 f32): lane<16 -> K = kk*4+{0,1}; lane>=16 -> K = kk*4+{2,3}.
    const int pos = tileBase + l16;
    const long long row = (long long)__builtin_nontemporal_load(idx + pos);
    const float* w1row = W1 + row * RANK;

    v2f a[8];
    #pragma unroll
    for (int kk = 0; kk < 8; ++kk) {
        a[kk] = *(const v2f*)(w1row + kk * 4 + half * 2);   // global_load_b64 (RT: keep in L2)
    }

    // 8 output column tiles of 16
    for (int ct = 0; ct < 8; ++ct) {
        const int n = ct * 16 + l16;
        const float* bt = &w2t[n * RANK];

        v8f c = {};
        #pragma unroll
        for (int kk = 0; kk < 8; ++kk) {
            // B-fragment (4x16 f32): VGPR v holds K=kbase+v, N=l16; lane half selects K+2.
            v2f b = *(const v2f*)(bt + kk * 4 + half * 2);   // ds_load_b64
            c = __builtin_amdgcn_wmma_f32_16x16x4_f32(
                    /*neg_a=*/false, a[kk],
                    /*neg_b=*/false, b,
                    /*c_mod=*/(short)0, c,
                    /*reuse_a=*/false, /*reuse_b=*/false);
        }

        // C/D layout: VGPR v -> M=v (lanes 0-15) / M=v+8 (lanes 16-31), N=l16.
        float* op = out + (long long)tileBase * EMB_DIM + ct * 16;
        #pragma unroll
        for (int v = 0; v < 8; ++v) {
            const int m = v + half * 8;
            // write-once stream: NT store, don't pollute L2 (keep W1 resident)
            __builtin_nontemporal_store(c[v], op + (long long)m * EMB_DIM + l16);
        }
    }
}

extern "C" void kernel_launch(void* const* d_in, const int* in_sizes, int n_in,
                              void* d_out, int out_size, void* d_ws, size_t ws_size,
                              hipStream_t stream) {
    const int*   idx = (const int*)  d_in[0];   // input indices [4096*200]
    const float* W1  = (const float*)d_in[1];   // [1e6, 32]
    const float* W2  = (const float*)d_in[2];   // [32, 128]
    float*       out = (float*)d_out;           // [4096*200, 128]

    const int n_pos  = in_sizes[0];             // 819200
    const int perBlk = 8 * 16;                  // 8 waves x 16 lookups
    const int blocks = (n_pos + perBlk - 1) / perBlk;

    LowRankEmbedding_19713899888995_kernel<<<blocks, 256, 0, stream>>>(
        idx, W1, W2, out, n_pos);
}